// RegionGraphAttention_1486058684820
// MI455X (gfx1250) — compile-verified
//
#include <hip/hip_runtime.h>
#include <math.h>

#define NN   100000
#define DD   128
#define HIDD 128
#define NHEAD 4
#define HDIM  32

typedef __attribute__((ext_vector_type(2))) float v2f;
typedef __attribute__((ext_vector_type(8))) float v8f;

// D = A(16x4,f32) * B(4x16,f32) + C(16x16,f32), full-precision CDNA5 WMMA
__device__ __forceinline__ v8f wmma_f32(v2f a, v2f b, v8f c) {
  return __builtin_amdgcn_wmma_f32_16x16x4_f32(
      /*neg_a=*/false, a, /*neg_b=*/false, b,
      /*c_mod=*/(short)0, c, /*reuse_a=*/false, /*reuse_b=*/false);
}

// order-preserving float <-> uint mapping for atomicMax on signed floats
__device__ __forceinline__ unsigned f2ord(float f) {
  unsigned u = __float_as_uint(f);
  return (u & 0x80000000u) ? ~u : (u | 0x80000000u);
}
__device__ __forceinline__ float ord2f(unsigned u) {
  unsigned b = (u & 0x80000000u) ? (u ^ 0x80000000u) : ~u;
  return __uint_as_float(b);
}

// ---------------- init: agg=0, sumexp=0, max=-1e9 ----------------
__global__ __launch_bounds__(256) void rga_init(unsigned* __restrict__ maxu,
                                                float* __restrict__ sumexp,
                                                float* __restrict__ agg) {
  size_t i = (size_t)blockIdx.x * 256 + threadIdx.x;
  if (i < (size_t)NN * HIDD) agg[i] = 0.0f;
  if (i < (size_t)NN * NHEAD) { maxu[i] = f2ord(-1e9f); sumexp[i] = 0.0f; }
}

// ---------------- fused QKV projection (WMMA f32) ----------------
// block = 256 threads = 8 waves; wave w owns output columns [16w,16w+16)
// of a 16-row node tile; one A-fragment feeds 3 WMMAs (Q,K,V).
__global__ __launch_bounds__(256) void rga_qkv(
    const float* __restrict__ X,
    const float* __restrict__ Wq, const float* __restrict__ bq,
    const float* __restrict__ Wk, const float* __restrict__ bk,
    const float* __restrict__ Wv, const float* __restrict__ bv,
    float* __restrict__ Q, float* __restrict__ K, float* __restrict__ Vv) {
  const int tile = blockIdx.x;            // 16 node rows per tile
  const int wave = threadIdx.x >> 5;
  const int lane = threadIdx.x & 31;
  const int m    = lane & 15;             // A row within tile
  const int koff = (lane >> 4) << 1;      // lanes 16-31 hold K=2,3
  const int col  = wave * 16 + m;         // B/C column

  const float* xrow = X + (size_t)(tile * 16 + m) * DD;

  v8f cq = {}; v8f ck = {}; v8f cv = {};
  for (int kk = 0; kk < DD; kk += 4) {
    const int k0 = kk + koff;
    v2f a, wq2, wk2, wv2;
    a.x = xrow[k0];
    a.y = xrow[k0 + 1];
    const size_t w0 = (size_t)k0 * HIDD + col;
    const size_t w1 = w0 + HIDD;
    wq2.x = Wq[w0]; wq2.y = Wq[w1];
    wk2.x = Wk[w0]; wk2.y = Wk[w1];
    wv2.x = Wv[w0]; wv2.y = Wv[w1];
    cq = wmma_f32(a, wq2, cq);
    ck = wmma_f32(a, wk2, ck);
    cv = wmma_f32(a, wv2, cv);
  }
  const float biq = bq[col], bik = bk[col], biv = bv[col];
  const int mrow = (lane >> 4) << 3;      // C: lanes 16-31 hold M=8..15
  for (int v = 0; v < 8; ++v) {
    const size_t r = (size_t)(tile * 16 + v + mrow) * HIDD + col;
    Q[r]  = cq[v] + biq;
    K[r]  = ck[v] + bik;
    Vv[r] = cv[v] + biv;
  }
}

// ---------------- per-edge attention scores + scatter max ----------------
// one wave per edge: lane/8 = head, lane%8 covers 4 of the 32 head dims
__global__ __launch_bounds__(256) void rga_score(
    const float* __restrict__ Q, const float* __restrict__ K,
    const int* __restrict__ ei, const float* __restrict__ ew,
    const float* __restrict__ We, const float* __restrict__ be,
    float* __restrict__ scores, unsigned* __restrict__ maxu, int E) {
  const int e = blockIdx.x * 8 + (threadIdx.x >> 5);
  if (e >= E) return;
  const int lane = threadIdx.x & 31;
  const int h    = lane >> 3;
  const int sub  = lane & 7;
  const int src  = ei[e];
  const int tgt  = ei[E + e];

  const float4 qv = *((const float4*)(Q + (size_t)tgt * HIDD + h * HDIM) + sub);
  const float4 kv = *((const float4*)(K + (size_t)src * HIDD + h * HDIM) + sub);
  float p = qv.x * kv.x + qv.y * kv.y + qv.z * kv.z + qv.w * kv.w;
  p += __shfl_xor(p, 1, 32);
  p += __shfl_xor(p, 2, 32);
  p += __shfl_xor(p, 4, 32);
  if (sub == 0) {
    const float s = p * 0.17677669529663687f   // 1/sqrt(32)
                  + ew[e] * We[h] + be[h];
    scores[(size_t)e * NHEAD + h] = s;
    atomicMax(&maxu[(size_t)tgt * NHEAD + h], f2ord(s));
  }
}

// ---------------- exp + scatter-sum + unnormalized aggregation ----------------
__global__ __launch_bounds__(256) void rga_agg(
    const float* __restrict__ Vv, const int* __restrict__ ei,
    const float* __restrict__ scores, const unsigned* __restrict__ maxu,
    float* __restrict__ sumexp, float* __restrict__ agg, int E) {
  const int e = blockIdx.x * 8 + (threadIdx.x >> 5);
  if (e >= E) return;
  const int lane = threadIdx.x & 31;
  const int h    = lane >> 3;
  const int sub  = lane & 7;
  const int src  = ei[e];
  const int tgt  = ei[E + e];

  const float s  = scores[(size_t)e * NHEAD + h];
  const float mx = ord2f(maxu[(size_t)tgt * NHEAD + h]);
  const float p  = __expf(s - mx);
  if (sub == 0) atomicAdd(&sumexp[(size_t)tgt * NHEAD + h], p);

  const float4 vv = *((const float4*)(Vv + (size_t)src * HIDD + h * HDIM) + sub);
  float* dst = agg + (size_t)tgt * HIDD + h * HDIM + sub * 4;
  atomicAdd(dst + 0, p * vv.x);
  atomicAdd(dst + 1, p * vv.y);
  atomicAdd(dst + 2, p * vv.z);
  atomicAdd(dst + 3, p * vv.w);
}

// ---------------- output projection (WMMA f32), normalize on load ----------------
__global__ __launch_bounds__(256) void rga_out(
    const float* __restrict__ agg, const float* __restrict__ sumexp,
    const float* __restrict__ Wo, const float* __restrict__ bo,
    float* __restrict__ out) {
  const int tile = blockIdx.x;
  const int wave = threadIdx.x >> 5;
  const int lane = threadIdx.x & 31;
  const int m    = lane & 15;
  const int koff = (lane >> 4) << 1;
  const int col  = wave * 16 + m;

  const float* arow = agg + (size_t)(tile * 16 + m) * HIDD;
  const float* srow = sumexp + (size_t)(tile * 16 + m) * NHEAD;

  v8f c = {};
  for (int kk = 0; kk < HIDD; kk += 4) {
    const int k0 = kk + koff;
    v2f a, b;
    a.x = arow[k0]     / (srow[k0 >> 5] + 1e-8f);
    a.y = arow[k0 + 1] / (srow[(k0 + 1) >> 5] + 1e-8f);
    b.x = Wo[(size_t)k0 * HIDD + col];
    b.y = Wo[(size_t)(k0 + 1) * HIDD + col];
    c = wmma_f32(a, b, c);
  }
  const float bias = bo[col];
  const int mrow = (lane >> 4) << 3;
  for (int v = 0; v < 8; ++v) {
    out[(size_t)(tile * 16 + v + mrow) * HIDD + col] = c[v] + bias;
  }
}

extern "C" void kernel_launch(void* const* d_in, const int* in_sizes, int n_in,
                              void* d_out, int out_size, void* d_ws, size_t ws_size,
                              hipStream_t stream) {
  (void)n_in; (void)out_size; (void)ws_size;
  const float* X  = (const float*)d_in[0];
  const int*   ei = (const int*)  d_in[1];
  const float* ew = (const float*)d_in[2];
  const float* Wq = (const float*)d_in[3];
  const float* bq = (const float*)d_in[4];
  const float* Wk = (const float*)d_in[5];
  const float* bk = (const float*)d_in[6];
  const float* Wv = (const float*)d_in[7];
  const float* bv = (const float*)d_in[8];
  const float* Wo = (const float*)d_in[9];
  const float* bo = (const float*)d_in[10];
  const float* We = (const float*)d_in[11];
  const float* be = (const float*)d_in[12];
  float* out = (float*)d_out;
  const int E = in_sizes[2];   // edge_weights element count

  // workspace layout
  float* Q      = (float*)d_ws;                       // N*128
  float* K      = Q      + (size_t)NN * HIDD;         // N*128
  float* Vv     = K      + (size_t)NN * HIDD;         // N*128
  float* agg    = Vv     + (size_t)NN * HIDD;         // N*128
  float* scores = agg    + (size_t)NN * HIDD;         // E*4
  float* sumexp = scores + (size_t)E * NHEAD;         // N*4
  unsigned* maxu = (unsigned*)(sumexp + (size_t)NN * NHEAD); // N*4

  rga_init <<<(NN * HIDD + 255) / 256, 256, 0, stream>>>(maxu, sumexp, agg);
  rga_qkv  <<<NN / 16,            256, 0, stream>>>(X, Wq, bq, Wk, bk, Wv, bv, Q, K, Vv);
  rga_score<<<(E + 7) / 8,        256, 0, stream>>>(Q, K, ei, ew, We, be, scores, maxu, E);
  rga_agg  <<<(E + 7) / 8,        256, 0, stream>>>(Vv, ei, scores, maxu, sumexp, agg, E);
  rga_out  <<<NN / 16,            256, 0, stream>>>(agg, sumexp, Wo, bo, out);
}